// DGCNN_cls_7206955123075
// MI455X (gfx1250) — compile-verified
//
#include <hip/hip_runtime.h>
#include <hip/hip_bf16.h>

typedef float v2f __attribute__((ext_vector_type(2)));
typedef float v8f __attribute__((ext_vector_type(8)));

#define NPTS 1024
#define BATCH 8
#define KNN 20
#define EPSV 1e-5f
#define NEG_INF (-3.4e38f)

static __device__ __forceinline__ v8f wmma4(v2f a, v2f b, v8f c) {
    return __builtin_amdgcn_wmma_f32_16x16x4_f32(
        false, a, false, b, (short)0, c, false, false);
}
static __device__ __forceinline__ v2f ld2(const float* p) {
    return *(const v2f*)p;
}

// ---------------------------------------------------------------------------
// x (B,3,N) -> h0 (B*N, 4) with padded zero 4th column
// ---------------------------------------------------------------------------
__global__ void transpose_kernel(const float* __restrict__ x, float* __restrict__ h0) {
    int i = blockIdx.x * blockDim.x + threadIdx.x;
    if (i >= BATCH * NPTS) return;
    int b = i / NPTS, n = i % NPTS;
    const float* xb = x + (size_t)b * 3 * NPTS;
    h0[(size_t)i * 4 + 0] = xb[0 * NPTS + n];
    h0[(size_t)i * 4 + 1] = xb[1 * NPTS + n];
    h0[(size_t)i * 4 + 2] = xb[2 * NPTS + n];
    h0[(size_t)i * 4 + 3] = 0.0f;
}

// ---------------------------------------------------------------------------
// xx[i] = sum_c H[i*ldh + c]^2
// ---------------------------------------------------------------------------
__global__ void xx_kernel(const float* __restrict__ H, int ldh, int C,
                          float* __restrict__ xx) {
    int i = blockIdx.x * blockDim.x + threadIdx.x;
    if (i >= BATCH * NPTS) return;
    float s = 0.0f;
    for (int c = 0; c < C; ++c) {
        float v = H[(size_t)i * ldh + c];
        s += v * v;
    }
    xx[i] = s;
}

// ---------------------------------------------------------------------------
// Pack one half of an EdgeConv weight (O x 2C, row ld = ldw) into a
// zero-padded contiguous (O x K4) buffer.
// ---------------------------------------------------------------------------
__global__ void pack_w_kernel(const float* __restrict__ w, int ldw, int off,
                              int C, int K4, int total, float* __restrict__ out) {
    int i = blockIdx.x * blockDim.x + threadIdx.x;
    if (i >= total) return;
    int o = i / K4, k = i % K4;
    out[i] = (k < C) ? w[(size_t)o * ldw + off + k] : 0.0f;
}

// ---------------------------------------------------------------------------
// Pairwise ranking matrix via WMMA:  pd[b][n][m] = 2 * <h_n, h_m> - xx[m]
// Each wave: one 16(M) x 64(N) strip = 1 A tile x NT B tiles.
// grid: (N/64, N/(16*NT), B), block (32,4)
// ---------------------------------------------------------------------------
template <int K, int NT>
__global__ void pair_wmma_kernel(const float* __restrict__ H, int ldh,
                                 const float* __restrict__ xx,
                                 float* __restrict__ pd) {
    int lane  = threadIdx.x;
    int tileM = blockIdx.x * 4 + threadIdx.y;
    int tileN0 = blockIdx.y * NT;
    int b     = blockIdx.z;
    const float* Hb  = H  + (size_t)b * NPTS * ldh;
    const float* xxb = xx + (size_t)b * NPTS;
    float* pdb = pd + (size_t)b * NPTS * NPTS;

    int rsel = lane & 15;
    int half = lane >> 4;
    int arow = tileM * 16 + rsel;
    int cols[NT];
#pragma unroll
    for (int nt = 0; nt < NT; ++nt) cols[nt] = (tileN0 + nt) * 16 + rsel;

    v8f acc[NT];
#pragma unroll
    for (int nt = 0; nt < NT; ++nt) acc[nt] = (v8f){};

#pragma unroll 8
    for (int k0 = 0; k0 < K; k0 += 4) {
        int ka = k0 + half * 2;
        v2f a = ld2(&Hb[(size_t)arow * ldh + ka]);
#pragma unroll
        for (int nt = 0; nt < NT; ++nt) {
            v2f bb = ld2(&Hb[(size_t)cols[nt] * ldh + ka]);
            acc[nt] = wmma4(a, bb, acc[nt]);
        }
    }
#pragma unroll
    for (int nt = 0; nt < NT; ++nt) {
        float xm = xxb[cols[nt]];
#pragma unroll
        for (int r = 0; r < 8; ++r) {
            int row = tileM * 16 + r + half * 8;
            pdb[(size_t)row * NPTS + cols[nt]] = 2.0f * acc[nt][r] - xm;
        }
    }
}

// ---------------------------------------------------------------------------
// GEMM via WMMA with pre-packed weights: Y[b][n][o] = sum_k A[n][k]*Wp[o][k]
// Wp is (O x K) contiguous zero-padded.  Each wave: 16 x (16*NT) strip.
// grid: (N/64, O/(16*NT), B), block (32,4)
// ---------------------------------------------------------------------------
template <int K, int NT>
__global__ void gemm_wmma_kernel(const float* __restrict__ A, int lda,
                                 const float* __restrict__ Wp,
                                 float* __restrict__ Y, int ldy,
                                 const float* __restrict__ bn, int act) {
    int lane  = threadIdx.x;
    int tileM = blockIdx.x * 4 + threadIdx.y;
    int tileN0 = blockIdx.y * NT;
    int b     = blockIdx.z;
    const float* Ab = A + (size_t)b * NPTS * lda;
    float*       Yb = Y + (size_t)b * NPTS * ldy;

    int rsel = lane & 15;
    int half = lane >> 4;
    int arow = tileM * 16 + rsel;
    int cols[NT];
#pragma unroll
    for (int nt = 0; nt < NT; ++nt) cols[nt] = (tileN0 + nt) * 16 + rsel;

    v8f acc[NT];
#pragma unroll
    for (int nt = 0; nt < NT; ++nt) acc[nt] = (v8f){};

#pragma unroll 8
    for (int k0 = 0; k0 < K; k0 += 4) {
        int ka = k0 + half * 2;
        v2f a = ld2(&Ab[(size_t)arow * lda + ka]);
#pragma unroll
        for (int nt = 0; nt < NT; ++nt) {
            v2f bb = ld2(&Wp[(size_t)cols[nt] * K + ka]);
            acc[nt] = wmma4(a, bb, acc[nt]);
        }
    }
#pragma unroll
    for (int nt = 0; nt < NT; ++nt) {
        int col = cols[nt];
        float s = 1.0f, t = 0.0f;
        if (bn) {
            int O = ldy;
            float gg = bn[col], be = bn[O + col], mm = bn[2 * O + col], vv = bn[3 * O + col];
            s = gg * rsqrtf(vv + EPSV);
            t = be - mm * s;
        }
#pragma unroll
        for (int r = 0; r < 8; ++r) {
            int row = tileM * 16 + r + half * 8;
            float v = acc[nt][r];
            if (bn) v = v * s + t;
            if (act) v = (v >= 0.0f) ? v : 0.2f * v;
            Yb[(size_t)row * ldy + col] = v;
        }
    }
}

// ---------------------------------------------------------------------------
// top-20 per row of pd (length 1024), one wave32 per row, register resident.
// Lane owns a contiguous 32-float chunk, loaded as 8 float4s.
// Ties broken toward lower index (matches jax.lax.top_k).
// grid: B*N/8 blocks, block (32,8)
// ---------------------------------------------------------------------------
__global__ void topk_kernel(const float* __restrict__ pd, int* __restrict__ idx) {
    int lane = threadIdx.x;
    int row  = blockIdx.x * 8 + threadIdx.y;
    const float4* rp4 = (const float4*)(pd + (size_t)row * NPTS);

    float vals[32];
#pragma unroll
    for (int j = 0; j < 8; ++j) {
        float4 q = rp4[lane * 8 + j];
        vals[4 * j + 0] = q.x;
        vals[4 * j + 1] = q.y;
        vals[4 * j + 2] = q.z;
        vals[4 * j + 3] = q.w;
    }

    unsigned consumed = 0;
    for (int it = 0; it < KNN; ++it) {
        float bv = NEG_INF;
        int   bi = 0x7fffffff;
#pragma unroll
        for (int s = 0; s < 32; ++s) {
            if (!((consumed >> s) & 1u)) {
                float v = vals[s];
                int gi = lane * 32 + s;
                if (v > bv || (v == bv && gi < bi)) { bv = v; bi = gi; }
            }
        }
#pragma unroll
        for (int off = 16; off >= 1; off >>= 1) {
            float ov = __shfl_xor(bv, off, 32);
            int   oi = __shfl_xor(bi, off, 32);
            if (ov > bv || (ov == bv && oi < bi)) { bv = ov; bi = oi; }
        }
        if (lane == 0) idx[(size_t)row * KNN + it] = bi;
        if (lane == (bi >> 5)) consumed |= 1u << (bi & 31);
    }
}

// ---------------------------------------------------------------------------
// EdgeConv epilogue: out(b,n,o) = max_k leaky( (z[b,idx,o] + c[b,n,o])*s + t )
// grid: B*N blocks, blockDim = O
// ---------------------------------------------------------------------------
__global__ void gathermax_kernel(const float* __restrict__ z,
                                 const float* __restrict__ cc,
                                 const int* __restrict__ idx,
                                 const float* __restrict__ bn, int O,
                                 float* __restrict__ hcOut) {
    int bnidx = blockIdx.x;
    int o = threadIdx.x;
    int b = bnidx / NPTS;

    float gg = bn[o], be = bn[O + o], mm = bn[2 * O + o], vv = bn[3 * O + o];
    float s = gg * rsqrtf(vv + EPSV);
    float t = be - mm * s;

    float ccv = cc[(size_t)bnidx * O + o];
    const int* ix = idx + (size_t)bnidx * KNN;
    size_t zb = (size_t)b * NPTS * O;

    float best = NEG_INF;
#pragma unroll
    for (int k = 0; k < KNN; ++k) {
        int j = ix[k];
        float v = (z[zb + (size_t)j * O + o] + ccv) * s + t;
        v = (v >= 0.0f) ? v : 0.2f * v;
        best = fmaxf(best, v);
    }
    hcOut[(size_t)bnidx * 512 + o] = best;
}

// ---------------------------------------------------------------------------
// Global max+mean pooling over N: h5 (B,N,128) -> g (B,256) [max | mean]
// ---------------------------------------------------------------------------
__global__ void pool_kernel(const float* __restrict__ h5, float* __restrict__ g) {
    int b = blockIdx.x, c = threadIdx.x;
    float mx = NEG_INF, sm = 0.0f;
    for (int n = 0; n < NPTS; ++n) {
        float v = h5[((size_t)b * NPTS + n) * 128 + c];
        mx = fmaxf(mx, v);
        sm += v;
    }
    g[(size_t)b * 256 + c]       = mx;
    g[(size_t)b * 256 + 128 + c] = sm * (1.0f / NPTS);
}

// ---------------------------------------------------------------------------
// FC head, one block per batch element.
// ---------------------------------------------------------------------------
__global__ void head_kernel(const float* __restrict__ g,
                            const float* __restrict__ l1w, const float* __restrict__ bn6,
                            const float* __restrict__ l2w, const float* __restrict__ l2b,
                            const float* __restrict__ bn7,
                            const float* __restrict__ l3w, const float* __restrict__ l3b,
                            const float* __restrict__ l4w, const float* __restrict__ l4b,
                            const float* __restrict__ l5w, const float* __restrict__ l5b,
                            float* __restrict__ out) {
    __shared__ float bufA[512];
    __shared__ float bufB[512];
    int b = blockIdx.x, tid = threadIdx.x;

    for (int i = tid; i < 256; i += 128) bufA[i] = g[(size_t)b * 256 + i];
    __syncthreads();

    for (int o = tid; o < 512; o += 128) {
        float acc = 0.0f;
        for (int i = 0; i < 256; ++i) acc += bufA[i] * l1w[(size_t)o * 256 + i];
        float s = bn6[o] * rsqrtf(bn6[1536 + o] + EPSV);
        float v = (acc - bn6[1024 + o]) * s + bn6[512 + o];
        bufB[o] = (v >= 0.0f) ? v : 0.2f * v;
    }
    __syncthreads();

    for (int o = tid; o < 256; o += 128) {
        float acc = l2b[o];
        for (int i = 0; i < 512; ++i) acc += bufB[i] * l2w[(size_t)o * 512 + i];
        float s = bn7[o] * rsqrtf(bn7[768 + o] + EPSV);
        float v = (acc - bn7[512 + o]) * s + bn7[256 + o];
        bufA[o] = (v >= 0.0f) ? v : 0.2f * v;
    }
    __syncthreads();

    for (int o = tid; o < 128; o += 128) {
        float acc = l3b[o];
        for (int i = 0; i < 256; ++i) acc += bufA[i] * l3w[(size_t)o * 256 + i];
        bufB[o] = acc;
    }
    __syncthreads();

    if (tid < 32) {
        float acc = l4b[tid];
        for (int i = 0; i < 128; ++i) acc += bufB[i] * l4w[(size_t)tid * 128 + i];
        bufA[tid] = acc;
    }
    __syncthreads();

    if (tid < 40) {
        float acc = l5b[tid];
        for (int i = 0; i < 32; ++i) acc += bufA[i] * l5w[(size_t)tid * 32 + i];
        out[(size_t)b * 40 + tid] = acc;
    }
}

// ---------------------------------------------------------------------------
extern "C" void kernel_launch(void* const* d_in, const int* in_sizes, int n_in,
                              void* d_out, int out_size, void* d_ws, size_t ws_size,
                              hipStream_t stream) {
    (void)in_sizes; (void)n_in; (void)out_size; (void)ws_size;

    const float* x   = (const float*)d_in[0];
    const float* w1  = (const float*)d_in[1];
    const float* bn1 = (const float*)d_in[2];
    const float* w2  = (const float*)d_in[3];
    const float* bn2 = (const float*)d_in[4];
    const float* w3  = (const float*)d_in[5];
    const float* bn3 = (const float*)d_in[6];
    const float* w4  = (const float*)d_in[7];
    const float* bn4 = (const float*)d_in[8];
    const float* w5  = (const float*)d_in[9];
    const float* bn5 = (const float*)d_in[10];
    const float* l1w = (const float*)d_in[11];
    const float* bn6 = (const float*)d_in[12];
    const float* l2w = (const float*)d_in[13];
    const float* l2b = (const float*)d_in[14];
    const float* bn7 = (const float*)d_in[15];
    const float* l3w = (const float*)d_in[16];
    const float* l3b = (const float*)d_in[17];
    const float* l4w = (const float*)d_in[18];
    const float* l4b = (const float*)d_in[19];
    const float* l5w = (const float*)d_in[20];
    const float* l5b = (const float*)d_in[21];

    // workspace layout (floats)
    float* ws   = (float*)d_ws;
    float* h0   = ws;                        // 32768
    float* hc   = h0 + 32768;                // 4194304
    float* xx   = hc + 4194304;              // 8192
    float* pd   = xx + 8192;                 // 8388608
    int*   idxb = (int*)(pd + 8388608);      // 163840
    float* z    = (float*)(idxb + 163840);   // 2097152
    float* cc   = z + 2097152;               // 2097152
    float* h5b  = cc + 2097152;              // 1048576
    float* gbuf = h5b + 1048576;             // 2048
    float* wnp  = gbuf + 2048;               // 32768 (max 256x128)
    float* wcp  = wnp + 32768;               // 32768

    const int np = (BATCH * NPTS + 255) / 256;
    transpose_kernel<<<np, 256, 0, stream>>>(x, h0);

    dim3 wv(32, 4);
    dim3 pgrid(NPTS / 64, NPTS / 64, BATCH);      // pair: 16 x 64-col strips
    dim3 tkg(BATCH * NPTS / 8);
    dim3 tkb(32, 8);

    // ---------------- layer 1: C=3 (K4=4), O=64 ----------------
    {
        xx_kernel<<<np, 256, 0, stream>>>(h0, 4, 3, xx);
        pair_wmma_kernel<4, 4><<<pgrid, wv, 0, stream>>>(h0, 4, xx, pd);
        topk_kernel<<<tkg, tkb, 0, stream>>>(pd, idxb);
        pack_w_kernel<<<1, 256, 0, stream>>>(w1, 6, 0, 3, 4, 64 * 4, wnp);
        pack_w_kernel<<<1, 256, 0, stream>>>(w1, 6, 3, 3, 4, 64 * 4, wcp);
        gemm_wmma_kernel<4, 4><<<dim3(16, 1, BATCH), wv, 0, stream>>>(h0, 4, wnp, z, 64, nullptr, 0);
        gemm_wmma_kernel<4, 4><<<dim3(16, 1, BATCH), wv, 0, stream>>>(h0, 4, wcp, cc, 64, nullptr, 0);
        gathermax_kernel<<<dim3(BATCH * NPTS), dim3(64), 0, stream>>>(z, cc, idxb, bn1, 64, hc + 0);
    }
    // ---------------- layer 2: C=64, O=64 ----------------
    {
        xx_kernel<<<np, 256, 0, stream>>>(hc, 512, 64, xx);
        pair_wmma_kernel<64, 4><<<pgrid, wv, 0, stream>>>(hc, 512, xx, pd);
        topk_kernel<<<tkg, tkb, 0, stream>>>(pd, idxb);
        pack_w_kernel<<<16, 256, 0, stream>>>(w2, 128, 0, 64, 64, 64 * 64, wnp);
        pack_w_kernel<<<16, 256, 0, stream>>>(w2, 128, 64, 64, 64, 64 * 64, wcp);
        gemm_wmma_kernel<64, 4><<<dim3(16, 1, BATCH), wv, 0, stream>>>(hc, 512, wnp, z, 64, nullptr, 0);
        gemm_wmma_kernel<64, 4><<<dim3(16, 1, BATCH), wv, 0, stream>>>(hc, 512, wcp, cc, 64, nullptr, 0);
        gathermax_kernel<<<dim3(BATCH * NPTS), dim3(64), 0, stream>>>(z, cc, idxb, bn2, 64, hc + 64);
    }
    // ---------------- layer 3: C=64 (input h2 at col 64), O=128 ----------------
    {
        xx_kernel<<<np, 256, 0, stream>>>(hc + 64, 512, 64, xx);
        pair_wmma_kernel<64, 4><<<pgrid, wv, 0, stream>>>(hc + 64, 512, xx, pd);
        topk_kernel<<<tkg, tkb, 0, stream>>>(pd, idxb);
        pack_w_kernel<<<32, 256, 0, stream>>>(w3, 128, 0, 64, 64, 128 * 64, wnp);
        pack_w_kernel<<<32, 256, 0, stream>>>(w3, 128, 64, 64, 64, 128 * 64, wcp);
        gemm_wmma_kernel<64, 4><<<dim3(16, 2, BATCH), wv, 0, stream>>>(hc + 64, 512, wnp, z, 128, nullptr, 0);
        gemm_wmma_kernel<64, 4><<<dim3(16, 2, BATCH), wv, 0, stream>>>(hc + 64, 512, wcp, cc, 128, nullptr, 0);
        gathermax_kernel<<<dim3(BATCH * NPTS), dim3(128), 0, stream>>>(z, cc, idxb, bn3, 128, hc + 128);
    }
    // ---------------- layer 4: C=128 (input h3 at col 128), O=256 ----------------
    {
        xx_kernel<<<np, 256, 0, stream>>>(hc + 128, 512, 128, xx);
        pair_wmma_kernel<128, 4><<<pgrid, wv, 0, stream>>>(hc + 128, 512, xx, pd);
        topk_kernel<<<tkg, tkb, 0, stream>>>(pd, idxb);
        pack_w_kernel<<<128, 256, 0, stream>>>(w4, 256, 0, 128, 128, 256 * 128, wnp);
        pack_w_kernel<<<128, 256, 0, stream>>>(w4, 256, 128, 128, 128, 256 * 128, wcp);
        gemm_wmma_kernel<128, 4><<<dim3(16, 4, BATCH), wv, 0, stream>>>(hc + 128, 512, wnp, z, 256, nullptr, 0);
        gemm_wmma_kernel<128, 4><<<dim3(16, 4, BATCH), wv, 0, stream>>>(hc + 128, 512, wcp, cc, 256, nullptr, 0);
        gathermax_kernel<<<dim3(BATCH * NPTS), dim3(256), 0, stream>>>(z, cc, idxb, bn4, 256, hc + 256);
    }

    // h5 = leaky(bn5(hc @ w5^T)) : (B,N,512) x (512->128); w5 already packed
    gemm_wmma_kernel<512, 4><<<dim3(16, 2, BATCH), wv, 0, stream>>>(hc, 512, w5, h5b, 128, bn5, 1);

    pool_kernel<<<BATCH, 128, 0, stream>>>(h5b, gbuf);

    head_kernel<<<BATCH, 128, 0, stream>>>(gbuf, l1w, bn6, l2w, l2b, bn7,
                                           l3w, l3b, l4w, l4b, l5w, l5b,
                                           (float*)d_out);
}